// SO3Convolution_39230231282263
// MI455X (gfx1250) — compile-verified
//
#include <hip/hip_runtime.h>
#include <hip/hip_bf16.h>

// ---------------- problem constants ----------------
#define BATCH   4
#define F_IN    32
#define F_OUT   64
#define NF      32          // 2*B_IN == 2*B_OUT
#define L_TOT   5456
#define NGRID   12
#define TWO_PI  6.28318530717958647692f

typedef __attribute__((ext_vector_type(2))) float v2f;
typedef __attribute__((ext_vector_type(8))) float v8f;

__device__ __host__ inline int loff(int l) { return (4 * l * l * l - l) / 3; } // sum_{j<l}(2j+1)^2

// ---------------- stage 1: DFT over gamma (last axis), e^{-i} ----------------
// x: [rows=z*i*b*a=131072, 32] real -> X1: [rows, 32 bins] complex
__global__ void k_dft_gamma(const float* __restrict__ x,
                            float* __restrict__ Xre, float* __restrict__ Xim, int total) {
  __shared__ float ct[32], st[32];
  int t = threadIdx.x;
  if (t < 32) { float a = (TWO_PI / 32.0f) * t; ct[t] = __cosf(a); st[t] = __sinf(a); }
  __syncthreads();
  int idx = blockIdx.x * blockDim.x + t;
  if (idx >= total) return;
  int ng = idx & 31; int row = idx >> 5;
  const float* xr = x + (size_t)row * 32;
  float re = 0.f, im = 0.f;
  int p = 0;
  #pragma unroll
  for (int g = 0; g < 32; ++g) {
    float v = xr[g];
    re += v * ct[p];
    im -= v * st[p];
    p = (p + ng) & 31;
  }
  Xre[idx] = re; Xim[idx] = im;
}

// ---------------- stage 2: DFT over alpha (axis -2), e^{-i}, complex ----------------
// X1: [zib=4096, a=32, ng=32] -> X2: [zib, ma=32, ng=32]
__global__ void k_dft_alpha(const float* __restrict__ Xre, const float* __restrict__ Xim,
                            float* __restrict__ Yre, float* __restrict__ Yim, int total) {
  __shared__ float ct[32], st[32];
  int t = threadIdx.x;
  if (t < 32) { float a = (TWO_PI / 32.0f) * t; ct[t] = __cosf(a); st[t] = __sinf(a); }
  __syncthreads();
  int idx = blockIdx.x * blockDim.x + t;
  if (idx >= total) return;
  int ng = idx & 31; int ma = (idx >> 5) & 31; int zib = idx >> 10;
  int base = zib * 1024 + ng;
  float re = 0.f, im = 0.f;
  int p = 0;
  #pragma unroll
  for (int a = 0; a < 32; ++a) {
    float c = ct[p], s = st[p];
    float xr = Xre[base + a * 32], xi = Xim[base + a * 32];
    re += xr * c + xi * s;    // (xr+i xi)*(c - i s)
    im += xi * c - xr * s;
    p = (p + ma) & 31;
  }
  Yre[idx] = re; Yim[idx] = im;
}

// ---------------- stage 3: beta contraction with wigner_fwd ----------------
// xh[li, z, i] = sum_b wf[b, li] * X2[z,i,b, m&31, n&31]
__global__ void k_xh(const float* __restrict__ X2re, const float* __restrict__ X2im,
                     const float* __restrict__ wf,
                     float* __restrict__ xhre, float* __restrict__ xhim, int total) {
  int idx = blockIdx.x * blockDim.x + threadIdx.x;
  if (idx >= total) return;
  int li = idx >> 7; int zi = idx & 127;        // zi = z*32+i
  int l = 15;
  while (loff(l) > li) --l;
  int nsz = 2 * l + 1;
  int rem = li - loff(l);
  int m = rem / nsz - l;
  int k = rem % nsz - l;
  int ma = (m + 32) & 31, ng = (k + 32) & 31;
  int base = (zi * 32) * 1024 + ma * 32 + ng;   // X2 index: ((z*32+i)*32 + b)*1024 + ma*32 + ng
  float re = 0.f, im = 0.f;
  #pragma unroll 4
  for (int b = 0; b < 32; ++b) {
    float w = wf[b * L_TOT + li];
    re += w * X2re[base + b * 1024];
    im += w * X2im[base + b * 1024];
  }
  xhre[idx] = re; xhim[idx] = im;
}

// ---------------- stage 4: kernel grid FT: yh[li,i,o] = sum_j D[li,j]*kernel[i,o,j] ----------------
__global__ void k_yh(const float* __restrict__ Dre, const float* __restrict__ Dim,
                     const float* __restrict__ ker,
                     float* __restrict__ yhre, float* __restrict__ yhim, int total) {
  int idx = blockIdx.x * blockDim.x + threadIdx.x;
  if (idx >= total) return;
  int li = idx >> 11; int io = idx & 2047;
  float re = 0.f, im = 0.f;
  #pragma unroll
  for (int j = 0; j < NGRID; ++j) {
    float kk = ker[io * NGRID + j];
    re += Dre[li * NGRID + j] * kk;
    im += Dim[li * NGRID + j] * kk;
  }
  yhre[idx] = re; yhim[idx] = im;
}

// ---------------- stage 5: per-degree complex GEMM via V_WMMA_F32_16X16X4_F32 ----------------
// A[(m,z),(k,i)] = xh[offl+m*n+k][z][i] ; B[(k,i),(nn,o)] = yh[offl+k*n+nn][i][o]
// C[(m,z),(nn,o)] -> zh[offl+m*n+nn][z][o]
// One wave computes a 16x32 complex output tile (two 16x16 N-subtiles sharing A).
// N dimension (64n) is an exact multiple of 32 -> no N masking. M rows are clamped
// (out-of-range rows compute garbage on valid memory, masked at the store).
__global__ void k_so3mm(const float* __restrict__ Are, const float* __restrict__ Aim,
                        const float* __restrict__ Bre, const float* __restrict__ Bim,
                        float* __restrict__ Cre, float* __restrict__ Cim,
                        int n, int offl) {
  const int lane = threadIdx.x;            // 0..31, one wave per block
  const int tN = blockIdx.x;               // 32-wide N tile, count = 2n (exact)
  const int tM = blockIdx.y;
  const int Mrows = 4 * n;
  const int half = lane >> 4;              // selects K pair {0,1} vs {2,3}
  const int lid = lane & 15;
  const int hb = half * 2;

  // A fragment source row (clamped: dead rows are masked at the store)
  int gr = tM * 16 + lid;
  if (gr > Mrows - 1) gr = Mrows - 1;
  const int aOff = (offl + (gr >> 2) * n) * 128 + (gr & 3) * 32 + hb; // + k*128 + i0
  const float* aRe = Are + aOff;
  const float* aIm = Aim + aOff;

  // B fragment columns (exact, no clamp needed)
  const int gc0 = tN * 32 + lid;
  const int gc1 = gc0 + 16;
  const int cb0 = (gc0 >> 6) * 2048 + (gc0 & 63);
  const int cb1 = (gc1 >> 6) * 2048 + (gc1 & 63);
  const int bRowStride = n * 2048;                     // per k block
  const float* bReB = Bre + offl * 2048 + hb * 64;     // + k*bRowStride + i0*64 (+64 for .y)
  const float* bImB = Bim + offl * 2048 + hb * 64;

  v8f acRe0 = {}, acIm0 = {}, acRe1 = {}, acIm1 = {};
  for (int k = 0; k < n; ++k) {
    const float* aRek = aRe + k * 128;
    const float* aImk = aIm + k * 128;
    const float* bRek = bReB + k * bRowStride;
    const float* bImk = bImB + k * bRowStride;
    #pragma unroll
    for (int i0 = 0; i0 < 32; i0 += 4) {
      v2f ar = *(const v2f*)(aRek + i0);   // K-consecutive pair, 8B aligned
      v2f ai = *(const v2f*)(aImk + i0);
      const int r = i0 * 64;
      v2f br0, bi0, br1, bi1, nb0, nb1;
      br0.x = bRek[r + cb0];      br0.y = bRek[r + 64 + cb0];
      bi0.x = bImk[r + cb0];      bi0.y = bImk[r + 64 + cb0];
      br1.x = bRek[r + cb1];      br1.y = bRek[r + 64 + cb1];
      bi1.x = bImk[r + cb1];      bi1.y = bImk[r + 64 + cb1];
      nb0.x = -bi0.x; nb0.y = -bi0.y;
      nb1.x = -bi1.x; nb1.y = -bi1.y;

      acRe0 = __builtin_amdgcn_wmma_f32_16x16x4_f32(false, ar, false, br0, (short)0, acRe0, false, false);
      acRe0 = __builtin_amdgcn_wmma_f32_16x16x4_f32(false, ai, false, nb0, (short)0, acRe0, false, false);
      acIm0 = __builtin_amdgcn_wmma_f32_16x16x4_f32(false, ar, false, bi0, (short)0, acIm0, false, false);
      acIm0 = __builtin_amdgcn_wmma_f32_16x16x4_f32(false, ai, false, br0, (short)0, acIm0, false, false);
      acRe1 = __builtin_amdgcn_wmma_f32_16x16x4_f32(false, ar, false, br1, (short)0, acRe1, false, false);
      acRe1 = __builtin_amdgcn_wmma_f32_16x16x4_f32(false, ai, false, nb1, (short)0, acRe1, false, false);
      acIm1 = __builtin_amdgcn_wmma_f32_16x16x4_f32(false, ar, false, bi1, (short)0, acIm1, false, false);
      acIm1 = __builtin_amdgcn_wmma_f32_16x16x4_f32(false, ai, false, br1, (short)0, acIm1, false, false);
    }
  }

  // C/D layout: lanes 0-15 -> (M=e, N=lid); lanes 16-31 -> (M=e+8, N=lid)
  #pragma unroll
  for (int e = 0; e < 8; ++e) {
    int gro = tM * 16 + e + half * 8;
    if (gro < Mrows) {
      int rbase = offl + (gro >> 2) * n;
      int zpart = (gro & 3) * 64;
      int a0 = (rbase + (gc0 >> 6)) * 256 + zpart + (gc0 & 63);
      int a1 = (rbase + (gc1 >> 6)) * 256 + zpart + (gc1 & 63);
      Cre[a0] = acRe0[e]; Cim[a0] = acIm0[e];
      Cre[a1] = acRe1[e]; Cim[a1] = acIm1[e];
    }
  }
}

// ---------------- stage 6: gather (l,m,n)->freq bins with wigner_inv ----------------
// Zf[z,o,b,ma,ng] = sum_{l>=max(|m|,|n|)} wi[b,li]*zh[li,z,o]
__global__ void k_gather(const float* __restrict__ zhre, const float* __restrict__ zhim,
                         const float* __restrict__ wi,
                         float* __restrict__ Zfre, float* __restrict__ Zfim, int total) {
  int idx = blockIdx.x * blockDim.x + threadIdx.x;
  if (idx >= total) return;
  int ng = idx & 31, ma = (idx >> 5) & 31, b = (idx >> 10) & 31, zo = idx >> 15;
  int m  = (ma <= 15) ? ma : ma - 32;
  int nn = (ng <= 15) ? ng : ng - 32;
  int am = m < 0 ? -m : m, an = nn < 0 ? -nn : nn;
  float re = 0.f, im = 0.f;
  if (am <= 15 && an <= 15) {
    int l0 = am > an ? am : an;
    for (int l = l0; l < 16; ++l) {
      int nsz = 2 * l + 1;
      int li = loff(l) + (m + l) * nsz + (nn + l);
      float w = wi[b * L_TOT + li];
      re += w * zhre[li * 256 + zo];
      im += w * zhim[li * 256 + zo];
    }
  }
  Zfre[idx] = re; Zfim[idx] = im;
}

// ---------------- stage 7: inverse DFT over ng (last axis), e^{+i}, scale 1/32 ----------------
__global__ void k_idft_g(const float* __restrict__ Xre, const float* __restrict__ Xim,
                         float* __restrict__ Yre, float* __restrict__ Yim, int total) {
  __shared__ float ct[32], st[32];
  int t = threadIdx.x;
  if (t < 32) { float a = (TWO_PI / 32.0f) * t; ct[t] = __cosf(a); st[t] = __sinf(a); }
  __syncthreads();
  int idx = blockIdx.x * blockDim.x + t;
  if (idx >= total) return;
  int g = idx & 31; int base = (idx >> 5) * 32;
  float re = 0.f, im = 0.f;
  int p = 0;
  #pragma unroll
  for (int ng = 0; ng < 32; ++ng) {
    float c = ct[p], s = st[p];
    float xr = Xre[base + ng], xi = Xim[base + ng];
    re += xr * c - xi * s;    // (xr+i xi)*(c + i s)
    im += xr * s + xi * c;
    p = (p + g) & 31;
  }
  Yre[idx] = re * (1.0f / 32.0f);
  Yim[idx] = im * (1.0f / 32.0f);
}

// ---------------- stage 8: inverse DFT over ma (axis -2), real part + bias ----------------
__global__ void k_idft_a(const float* __restrict__ Xre, const float* __restrict__ Xim,
                         const float* __restrict__ bias,
                         float* __restrict__ out, int total) {
  __shared__ float ct[32], st[32];
  int t = threadIdx.x;
  if (t < 32) { float a = (TWO_PI / 32.0f) * t; ct[t] = __cosf(a); st[t] = __sinf(a); }
  __syncthreads();
  int idx = blockIdx.x * blockDim.x + t;
  if (idx >= total) return;
  int g = idx & 31; int a = (idx >> 5) & 31; int zob = idx >> 10;
  int o = (zob >> 5) & 63;
  int base = zob * 1024 + g;
  float re = 0.f;
  int p = 0;
  #pragma unroll
  for (int ma = 0; ma < 32; ++ma) {
    float c = ct[p], s = st[p];
    re += Xre[base + ma * 32] * c - Xim[base + ma * 32] * s;
    p = (p + a) & 31;
  }
  out[idx] = re * (1.0f / 32.0f) + bias[o];
}

// ---------------- host side ----------------
extern "C" void kernel_launch(void* const* d_in, const int* in_sizes, int n_in,
                              void* d_out, int out_size, void* d_ws, size_t ws_size,
                              hipStream_t stream) {
  const float* x    = (const float*)d_in[0];   // [4,32,32,32,32]
  const float* ker  = (const float*)d_in[1];   // [32,64,12]
  const float* bias = (const float*)d_in[2];   // [64]
  const float* wf   = (const float*)d_in[3];   // [32,5456]
  const float* wi   = (const float*)d_in[4];   // [32,5456]
  const float* Dre  = (const float*)d_in[5];   // [5456,12]
  const float* Dim  = (const float*)d_in[6];   // [5456,12]
  float* ws = (float*)d_ws;

  // workspace layout (floats), with aliasing of dead regions:
  const size_t NX   = 4194304;   // 4*32*32*32*32
  const size_t NXH  = 698368;    // 5456*128
  const size_t NYH  = 11173888;  // 5456*2048
  const size_t NZH  = 1396736;   // 5456*256
  const size_t NZF  = 8388608;   // 4*64*32*32*32

  float* X1re = ws;                         // region A (16,777,216 floats)
  float* X1im = X1re + NX;
  float* X2re = X1im + NX;
  float* X2im = X2re + NX;
  float* xhre = ws + 4 * NX;                // region B
  float* xhim = xhre + NXH;
  float* yhre = xhim + NXH;                 // region C (22,347,776 floats)
  float* yhim = yhre + NYH;
  float* zhre = yhim + NYH;                 // region D
  float* zhim = zhre + NZH;
  // aliases (regions A and C are dead by the time these are written):
  float* Zfre = ws;                         // = X1re..  (needs 16,777,216)
  float* Zfim = ws + NZF;
  float* Zgre = yhre;                       // = yh region (needs 16,777,216 of 22,347,776)
  float* Zgim = yhre + NZF;

  const int TB = 256;

  // stage 1+2: forward FFT2 as two DFT stages
  {
    int total = (int)NX;
    k_dft_gamma<<<(total + TB - 1) / TB, TB, 0, stream>>>(x, X1re, X1im, total);
    k_dft_alpha<<<(total + TB - 1) / TB, TB, 0, stream>>>(X1re, X1im, X2re, X2im, total);
  }
  // stage 3: xh
  {
    int total = (int)NXH;
    k_xh<<<(total + TB - 1) / TB, TB, 0, stream>>>(X2re, X2im, wf, xhre, xhim, total);
  }
  // stage 4: yh
  {
    int total = (int)NYH;
    k_yh<<<(total + TB - 1) / TB, TB, 0, stream>>>(Dre, Dim, ker, yhre, yhim, total);
  }
  // stage 5: per-degree complex GEMMs on the FP32 WMMA path
  for (int l = 0; l < 16; ++l) {
    int n = 2 * l + 1;
    dim3 grid(2 * n, (4 * n + 15) / 16);    // N tiles of 32 (exact), M tiles of 16
    k_so3mm<<<grid, 32, 0, stream>>>(xhre, xhim, yhre, yhim, zhre, zhim, n, loff(l));
  }
  // stage 6: gather into frequency grid
  {
    int total = (int)NZF;
    k_gather<<<(total + TB - 1) / TB, TB, 0, stream>>>(zhre, zhim, wi, Zfre, Zfim, total);
  }
  // stage 7+8: inverse FFT2 as two DFT stages; real part + bias
  {
    int total = (int)NZF;
    k_idft_g<<<(total + TB - 1) / TB, TB, 0, stream>>>(Zfre, Zfim, Zgre, Zgim, total);
    k_idft_a<<<(total + TB - 1) / TB, TB, 0, stream>>>(Zgre, Zgim, bias, (float*)d_out, total);
  }
}